// nconv_33122787787064
// MI455X (gfx1250) — compile-verified
//
#include <hip/hip_runtime.h>
#include <math.h>

typedef __attribute__((ext_vector_type(2))) float v2f;
typedef __attribute__((ext_vector_type(8))) float v8f;

// Problem dims (from reference): B,C,C1,N,T = 4,32,2,2000,24 ; mlp out = 10 ; topk = 5
#define Bn   4
#define Cn   32
#define C1n  2
#define Nn   2000
#define Tn   24
#define Dn   10
#define NP   2048            // padded N for GEMM (M and K both padded with zeros)
#define COLS (Cn * Tn)       // 768 = flattened (c,l) column dim of the batched GEMM

#define XOUT_SIZE (Bn * Cn * Nn * Tn)     // 6,144,000 floats, first part of d_out
#define INV_SQRT10 0.31622776601683794f

__device__ __forceinline__ float sigmoidf(float z) { return 1.0f / (1.0f + __expf(-z)); }

// ---------------------------------------------------------------------------
// K1: mp[n,b,v,d] = sum_t mask[n,b,v,t] * mlp_w[d,t] + mlp_b[d]
// ---------------------------------------------------------------------------
__global__ __launch_bounds__(256) void mp_kernel(const float* __restrict__ mask,
                                                 const float* __restrict__ mlp_w,
                                                 const float* __restrict__ mlp_b,
                                                 float* __restrict__ mp) {
  int idx = blockIdx.x * 256 + threadIdx.x;
  if (idx >= Bn * C1n * Nn * Dn) return;
  int d   = idx % Dn;
  int row = idx / Dn;                       // (n*C1 + b)*N + v
  const float* mrow = mask + (size_t)row * Tn;
  float acc = mlp_b[d];
  #pragma unroll
  for (int t = 0; t < Tn; ++t) acc += mrow[t] * mlp_w[d * Tn + t];
  mp[idx] = acc;
}

// ---------------------------------------------------------------------------
// K2: Asum[n][v][w] (padded to 2048x2048, zeros outside 2000x2000)
//     = 2*A[v,w] + 0.002 * sum_b sigmoid((mp_v . mp_w + 0.001*alpha)/sqrt(10))
//     alpha = alpha1[v,w] if w>v else alpha2[v,w]
// ---------------------------------------------------------------------------
__global__ __launch_bounds__(256) void asum_kernel(const float* __restrict__ A,
                                                   const float* __restrict__ mp,
                                                   const float* __restrict__ alpha1,
                                                   const float* __restrict__ alpha2,
                                                   float* __restrict__ Asum) {
  size_t idx = (size_t)blockIdx.x * 256 + threadIdx.x;   // over Bn*NP*NP
  int w = (int)(idx & (NP - 1));
  int v = (int)((idx >> 11) & (NP - 1));
  int n = (int)(idx >> 22);
  float r = 0.0f;
  if (v < Nn && w < Nn) {
    float a  = A[(size_t)v * Nn + w];
    float al = (w > v) ? alpha1[(size_t)v * Nn + w] : alpha2[(size_t)v * Nn + w];
    float s  = 0.0f;
    #pragma unroll
    for (int b = 0; b < C1n; ++b) {
      const float* mpv = mp + (((size_t)n * C1n + b) * Nn + v) * Dn;
      const float* mpw = mp + (((size_t)n * C1n + b) * Nn + w) * Dn;
      float dot = 0.0f;
      #pragma unroll
      for (int d = 0; d < Dn; ++d) dot += mpv[d] * mpw[d];
      s += sigmoidf((dot + 0.001f * al) * INV_SQRT10);
    }
    r = 2.0f * a + 0.002f * s;
  }
  Asum[idx] = r;
}

// ---------------------------------------------------------------------------
// K3: Xp[n][w][c*24+l] = x[n][c][w][l]  (w padded to 2048 with zeros)
// ---------------------------------------------------------------------------
__global__ __launch_bounds__(256) void xp_kernel(const float* __restrict__ x,
                                                 float* __restrict__ Xp) {
  size_t idx = (size_t)blockIdx.x * 256 + threadIdx.x;   // over Bn*NP*COLS
  int col = (int)(idx % COLS);
  int w   = (int)((idx / COLS) & (NP - 1));
  int n   = (int)(idx / ((size_t)COLS * NP));
  int c = col / Tn, l = col % Tn;
  float val = 0.0f;
  if (w < Nn) val = x[(((size_t)n * Cn + c) * Nn + w) * Tn + l];
  Xp[idx] = val;
}

// ---------------------------------------------------------------------------
// K4: top-5 per (n,b,v) row of Af (recomputed on the fly) + mask gather-max.
//     One wave32 per row. jax tie-break: larger value first, then lower index.
// ---------------------------------------------------------------------------
__global__ __launch_bounds__(128) void topk_kernel(const float* __restrict__ A,
                                                   const float* __restrict__ mask,
                                                   const float* __restrict__ mp,
                                                   const float* __restrict__ alpha1,
                                                   const float* __restrict__ alpha2,
                                                   float* __restrict__ mask_out) {
  int gtid = blockIdx.x * 128 + threadIdx.x;
  int wave = gtid >> 5;
  int lane = threadIdx.x & 31;
  if (wave >= Bn * C1n * Nn) return;        // grid is exact; uniform per wave
  int v = wave % Nn;
  int b = (wave / Nn) % C1n;
  int n = wave / (Nn * C1n);

  const float* mpb = mp + ((size_t)(n * C1n + b)) * Nn * Dn;
  float mpv[Dn];
  #pragma unroll
  for (int d = 0; d < Dn; ++d) mpv[d] = mpb[(size_t)v * Dn + d];

  float val[5];
  int   ids[5];
  #pragma unroll
  for (int j = 0; j < 5; ++j) { val[j] = -INFINITY; ids[j] = 0x7fffffff; }

  for (int w = lane; w < Nn; w += 32) {
    const float* mpw = mpb + (size_t)w * Dn;
    float dot = 0.0f;
    #pragma unroll
    for (int d = 0; d < Dn; ++d) dot += mpv[d] * mpw[d];
    float al = (w > v) ? alpha1[(size_t)v * Nn + w] : alpha2[(size_t)v * Nn + w];
    float af = A[(size_t)v * Nn + w] + 0.002f * sigmoidf((dot + 0.001f * al) * INV_SQRT10);
    // sorted insert into local top-5
    #pragma unroll
    for (int j = 0; j < 5; ++j) {
      if (af > val[j] || (af == val[j] && w < ids[j])) {
        #pragma unroll
        for (int q = 4; q > j; --q) { val[q] = val[q - 1]; ids[q] = ids[q - 1]; }
        val[j] = af; ids[j] = w;
        break;
      }
    }
  }

  // 5 rounds of cross-lane argmax (value desc, index asc tie-break)
  int head = 0;
  int sel[5];
  #pragma unroll
  for (int r = 0; r < 5; ++r) {
    float cv = (head < 5) ? val[head] : -INFINITY;
    int   ci = (head < 5) ? ids[head] : 0x7fffffff;
    float bv = cv; int bi = ci;
    #pragma unroll
    for (int off = 16; off > 0; off >>= 1) {
      float ov = __shfl_xor(bv, off, 32);
      int   oi = __shfl_xor(bi, off, 32);
      if (ov > bv || (ov == bv && oi < bi)) { bv = ov; bi = oi; }
    }
    sel[r] = bi;
    if (head < 5 && ci == bi && cv == bv) ++head;
  }

  if (lane < Tn) {
    const float* mb = mask + ((size_t)(n * C1n + b)) * Nn * Tn;
    float mo = mb[(size_t)sel[0] * Tn + lane];
    #pragma unroll
    for (int r = 1; r < 5; ++r) mo = fmaxf(mo, mb[(size_t)sel[r] * Tn + lane]);
    mask_out[((size_t)(n * C1n + b) * Nn + v) * Tn + lane] = mo;
  }
}

// ---------------------------------------------------------------------------
// K5: batched fp32 WMMA GEMM:  out[n][c][v][l] = sum_w Asum[n][v][w]*Xp[n][w][c*24+l]
//     Block: 256 thr (8 waves), tile 128(M) x 64(Ncols), K tiled by 32.
//     Tiles staged global->LDS with GLOBAL_LOAD_ASYNC_TO_LDS_B128 (ASYNCcnt),
//     compute with V_WMMA_F32_16X16X4_F32 (2x2 fragments per wave).
// ---------------------------------------------------------------------------
#define BM 128
#define BNc 64
#define BK 32
#define SA 36   // padded LDS stride for A rows (floats): 144B, 16B-aligned, conflict-free
#define SB 72   // padded LDS stride for B rows (floats): 288B, 16B-aligned, conflict-free

__device__ __forceinline__ void async_b128_to_lds(const float* gsrc, const float* lds_dst) {
  unsigned int       l = (unsigned int)(unsigned long long)(uintptr_t)lds_dst; // low 32b = LDS offset
  unsigned long long g = (unsigned long long)(uintptr_t)gsrc;
  asm volatile("global_load_async_to_lds_b128 %0, %1, off" :: "v"(l), "v"(g) : "memory");
}

__global__ __launch_bounds__(256) void gemm_kernel(const float* __restrict__ Asum,
                                                   const float* __restrict__ Xp,
                                                   float* __restrict__ out) {
  __shared__ float As[BM][SA];   // As[m][k], row-major tile of Asum
  __shared__ float Bs[BK][SB];   // Bs[k][n]

  int batch = blockIdx.z;
  int m0 = blockIdx.x * BM;
  int n0 = blockIdx.y * BNc;
  const float* Ab = Asum + (size_t)batch * NP * NP;
  const float* Xb = Xp   + (size_t)batch * NP * COLS;

  int tid  = threadIdx.x;
  int wave = tid >> 5;
  int lane = tid & 31;
  int wm = (wave & 3) * 32;        // 4 wave-rows
  int wn = (wave >> 2) * 32;       // 2 wave-cols
  int lm = lane & 15;              // M / Ncol within fragment
  int kh = (lane >> 4) << 1;       // 0 for lanes 0-15, 2 for lanes 16-31

  v8f acc[2][2] = {};

  for (int k0 = 0; k0 < NP; k0 += BK) {
    // --- async stage A tile: As[m][kq..kq+3] <= Ab[(m0+m)*NP + k0+kq..+3]
    #pragma unroll
    for (int i = tid; i < (BM * BK / 4); i += 256) {   // 4 iters/thread
      int m  = i >> 3;             // 8 x float4 per 32-wide k row
      int kq = (i & 7) << 2;
      async_b128_to_lds(Ab + (size_t)(m0 + m) * NP + k0 + kq, &As[m][kq]);
    }
    // --- async stage B tile: Bs[k][nq..nq+3] <= Xb[(k0+k)*COLS + n0+nq..+3]
    #pragma unroll
    for (int i = tid; i < (BK * BNc / 4); i += 256) {  // 2 iters/thread
      int k  = i >> 4;             // 16 x float4 per 64-wide row
      int nq = (i & 15) << 2;
      async_b128_to_lds(Xb + (size_t)(k0 + k) * COLS + n0 + nq, &Bs[k][nq]);
    }
    asm volatile("s_wait_asynccnt 0x0" ::: "memory");
    __syncthreads();

    #pragma unroll
    for (int kk = 0; kk < BK; kk += 4) {
      int ks = kk + kh;            // lanes 0-15 take k{0,1}, lanes 16-31 take k{2,3}
      #pragma unroll
      for (int mi = 0; mi < 2; ++mi) {
        const float* ap = &As[wm + mi * 16 + lm][ks];  // 2 consecutive floats: ds_load_b64
        v2f a; a.x = ap[0]; a.y = ap[1];
        #pragma unroll
        for (int ni = 0; ni < 2; ++ni) {
          v2f bfr;
          bfr.x = Bs[ks + 0][wn + ni * 16 + lm];
          bfr.y = Bs[ks + 1][wn + ni * 16 + lm];
          acc[mi][ni] = __builtin_amdgcn_wmma_f32_16x16x4_f32(
              false, a, false, bfr, (short)0, acc[mi][ni], false, false);
        }
      }
    }
    __syncthreads();
  }

  // Store: C/D layout — VGPR r: lanes 0-15 -> M=r, lanes 16-31 -> M=8+r; Ncol = lane%16
  int mhalf = (lane >> 4) << 3;
  #pragma unroll
  for (int mi = 0; mi < 2; ++mi) {
    #pragma unroll
    for (int ni = 0; ni < 2; ++ni) {
      int col = n0 + wn + ni * 16 + lm;
      int c = col / Tn, l = col % Tn;
      #pragma unroll
      for (int r = 0; r < 8; ++r) {
        int vrow = m0 + wm + mi * 16 + mhalf + r;
        if (vrow < Nn)
          out[(size_t)batch * Cn * Nn * Tn + (size_t)c * Nn * Tn + (size_t)vrow * Tn + l]
              = acc[mi][ni][r];
      }
    }
  }
}

// ---------------------------------------------------------------------------
// Launch. Workspace layout (floats): mp[160000] | Asum[4*2048*2048] | Xp[4*2048*768]
// total ~93 MB of d_ws.
// ---------------------------------------------------------------------------
extern "C" void kernel_launch(void* const* d_in, const int* in_sizes, int n_in,
                              void* d_out, int out_size, void* d_ws, size_t ws_size,
                              hipStream_t stream) {
  const float* x      = (const float*)d_in[0];
  const float* A      = (const float*)d_in[1];
  const float* mask   = (const float*)d_in[2];
  const float* mlp_w  = (const float*)d_in[3];
  const float* mlp_b  = (const float*)d_in[4];
  const float* alpha1 = (const float*)d_in[5];
  const float* alpha2 = (const float*)d_in[6];
  (void)in_sizes; (void)n_in; (void)out_size; (void)ws_size;

  float* out  = (float*)d_out;
  float* ws   = (float*)d_ws;
  float* mp   = ws;
  float* Asum = ws + 160000;
  float* Xp   = ws + 160000 + (size_t)Bn * NP * NP;

  mp_kernel<<<(Bn * C1n * Nn * Dn + 255) / 256, 256, 0, stream>>>(mask, mlp_w, mlp_b, mp);
  asum_kernel<<<(Bn * NP * NP) / 256, 256, 0, stream>>>(A, mp, alpha1, alpha2, Asum);
  xp_kernel<<<(Bn * NP * COLS) / 256, 256, 0, stream>>>(x, Xp);
  topk_kernel<<<(Bn * C1n * Nn) / 4, 128, 0, stream>>>(A, mask, mp, alpha1, alpha2,
                                                       out + XOUT_SIZE);
  dim3 g(NP / BM, COLS / BNc, Bn);
  gemm_kernel<<<g, 256, 0, stream>>>(Asum, Xp, out);
}